// SimplePagedKVCache_1322849927706
// MI455X (gfx1250) — compile-verified
//
#include <hip/hip_runtime.h>
#include <hip/hip_bf16.h>

// ---------------- problem constants (from reference config) ----------------
#define B_      8
#define L_      1024
#define BS_     16
#define MAXB_   128
#define H_      8
#define D_      128
#define NLAYERS_ 2
#define TOTAL_  (B_ * MAXB_ * NLAYERS_)   // 2048 physical blocks
#define CACHE_F4 ((long)TOTAL_ * H_ * BS_ * (D_ / 4))   // float4 per cache = 8,388,608

// true vector type matching the builtin's expected pointee (v4i)
typedef int v4i __attribute__((vector_size(16)));
typedef __attribute__((address_space(1))) v4i glb_v4i;   // global (as1) int4
typedef __attribute__((address_space(3))) v4i lds_v4i;   // LDS (as3) int4
typedef __attribute__((address_space(3))) void lds_void_t;

// ---------------- CDNA5 async global<->LDS helpers -------------------------
__device__ __forceinline__ void async_ld_b128(const float4* g, float4* l) {
#if __has_builtin(__builtin_amdgcn_global_load_async_to_lds_b128)
  __builtin_amdgcn_global_load_async_to_lds_b128(
      (glb_v4i*)g, (lds_v4i*)l, /*offset=*/0, /*cpol=*/0);
#else
  unsigned lo = (unsigned)(unsigned long)(lds_void_t*)l;
  asm volatile("global_load_async_to_lds_b128 %0, %1, off"
               :: "v"(lo), "v"((unsigned long long)g) : "memory");
#endif
}

__device__ __forceinline__ void async_st_b128(float4* g, const float4* l) {
#if __has_builtin(__builtin_amdgcn_global_store_async_from_lds_b128)
  __builtin_amdgcn_global_store_async_from_lds_b128(
      (glb_v4i*)g, (lds_v4i*)l, /*offset=*/0, /*cpol=*/0);
#else
  unsigned lo = (unsigned)(unsigned long)(lds_void_t*)l;
  asm volatile("global_store_async_from_lds_b128 %0, %1, off"
               :: "v"((unsigned long long)g), "v"(lo) : "memory");
#endif
}

__device__ __forceinline__ void wait_async0() {
#if __has_builtin(__builtin_amdgcn_s_wait_asynccnt)
  __builtin_amdgcn_s_wait_asynccnt(0);
#else
  asm volatile("s_wait_asynccnt 0" ::: "memory");
#endif
}

// ---------------- kernel 1: block allocation (reference cumsum math) -------
__global__ void __launch_bounds__(256)
alloc_bt_kernel(const int* __restrict__ input_len,
                const int* __restrict__ block_tables,
                const int* __restrict__ seq_lens,
                const int* __restrict__ free_blocks,
                int* __restrict__ bt_out) {
  __shared__ int s_old[B_], s_new[B_], s_start[B_];
  if (threadIdx.x == 0) {
    int cum = 0;
    for (int b = 0; b < B_; ++b) {
      int sl = seq_lens[b];
      int il = input_len[b];
      int oldnb = (sl + BS_ - 1) / BS_;
      int newnb = (sl + il + BS_ - 1) / BS_;
      cum += (newnb - oldnb);
      s_old[b] = oldnb;
      s_new[b] = newnb;
      s_start[b] = TOTAL_ - cum;   // num_free - cumsum(alloc)
    }
  }
  __syncthreads();
  for (int i = threadIdx.x; i < B_ * MAXB_; i += blockDim.x) {
    int b = i / MAXB_;
    int s = i % MAXB_;
    int v;
    if (s >= s_old[b] && s < s_new[b]) {
      int fi = s_start[b] + (s - s_old[b]);
      fi = fi < 0 ? 0 : (fi > TOTAL_ - 1 ? TOTAL_ - 1 : fi);  // jnp.clip
      v = free_blocks[fi];
    } else {
      v = block_tables[i];
    }
    bt_out[i] = v;
  }
}

// ---------------- kernel 2: bulk cache copy via async LDS DMA --------------
// Each wave32 owns a private double-buffered LDS region; per stage it issues
// UNR async b128 loads (512B * UNR), waits ASYNCcnt==0 (which also drains the
// previous stage's stores before that buffer is overwritten next iteration),
// then issues UNR async b128 stores straight out of LDS. No VGPR data path.
__global__ void __launch_bounds__(256)
copy_async_kernel(const float4* __restrict__ src, float4* __restrict__ dst, int n4) {
  constexpr int WAVES = 8;   // 256 threads / wave32
  constexpr int UNR   = 4;   // 4 x (32 lanes x 16B) = 2 KB per wave per stage
  __shared__ float4 buf[2][WAVES][UNR][32];   // 32 KB per workgroup

  const int wave = threadIdx.x >> 5;
  const int lane = threadIdx.x & 31;
  const long gwave  = (long)blockIdx.x * WAVES + wave;
  const long nwaves = (long)gridDim.x * WAVES;
  constexpr int CH = UNR * 32;                 // float4 per wave per stage

  int st = 0;
  for (long base = gwave * CH; base < n4; base += nwaves * CH, st ^= 1) {
#pragma unroll
    for (int u = 0; u < UNR; ++u) {
      long gi = base + u * 32 + lane;
      if (gi < n4) async_ld_b128(&src[gi], &buf[st][wave][u][lane]);
    }
    wait_async0();   // loads of this stage done; prior stage's stores drained
#pragma unroll
    for (int u = 0; u < UNR; ++u) {
      long gi = base + u * 32 + lane;
      if (gi < n4) async_st_b128(&dst[gi], &buf[st][wave][u][lane]);
    }
  }
  // trailing async stores drained by the implicit S_WAIT_IDLE at S_ENDPGM
}

// ---------------- kernel 3: scatter new K/V tokens into paged cache --------
// One wave per (token, head): 32 lanes x float4 = one 128-float head row.
// Coalesced 512B load from packed K/V, contiguous 512B store into the block.
__global__ void __launch_bounds__(256)
scatter_kernel(const float4* __restrict__ key_states,
               const float4* __restrict__ value_states,
               const int* __restrict__ cu_seqlens,
               const int* __restrict__ seq_lens,
               const int* __restrict__ bt,
               float4* __restrict__ out) {
  constexpr int D4 = D_ / 4;   // 32 float4 per row == one lane each
  const int wave = threadIdx.x >> 5;
  const int lane = threadIdx.x & 31;
  const int th = blockIdx.x * (blockDim.x >> 5) + wave;   // [0, B*L*H)
  if (th >= B_ * L_ * H_) return;

  const int h  = th % H_;
  const int tg = th / H_;          // [0, B*L)
  const int b  = tg / L_;
  const int t  = tg % L_;

  const int tok = cu_seqlens[b] + t;          // index into packed K/V
  const int pos = seq_lens[b] + t;            // absolute position in sequence
  const int blk = bt[b * MAXB_ + pos / BS_];  // physical block id
  const int off = pos % BS_;                  // offset within block

  const long srcIdx = ((long)tok * H_ + h) * D4 + lane;
  const long dstIdx = (((long)blk * H_ + h) * BS_ + off) * D4 + lane;

  out[dstIdx]            = key_states[srcIdx];    // k_new half
  out[CACHE_F4 + dstIdx] = value_states[srcIdx];  // v_new half
}

// ---------------- launcher --------------------------------------------------
extern "C" void kernel_launch(void* const* d_in, const int* in_sizes, int n_in,
                              void* d_out, int out_size, void* d_ws, size_t ws_size,
                              hipStream_t stream) {
  // input order: layer_idx, key_states, value_states, input_len, cu_seqlens,
  //              k_cache, v_cache, block_tables, seq_lens, free_blocks
  const float* key_states   = (const float*)d_in[1];
  const float* value_states = (const float*)d_in[2];
  const int*   input_len    = (const int*)d_in[3];
  const int*   cu_seqlens   = (const int*)d_in[4];
  const float* k_cache      = (const float*)d_in[5];
  const float* v_cache      = (const float*)d_in[6];
  const int*   block_tables = (const int*)d_in[7];
  const int*   seq_lens     = (const int*)d_in[8];
  const int*   free_blocks  = (const int*)d_in[9];

  float* out   = (float*)d_out;
  int*   bt_ws = (int*)d_ws;          // B*MAXB ints of scratch

  // 1) block allocation -> updated block table in workspace
  alloc_bt_kernel<<<1, 256, 0, stream>>>(input_len, block_tables, seq_lens,
                                         free_blocks, bt_ws);

  // 2) copy caches into output via async global->LDS->global DMA path
  const int n4 = (int)CACHE_F4;       // 8,388,608 float4 per cache
  const long vOffFloats = (long)TOTAL_ * H_ * BS_ * D_;   // start of v half
  copy_async_kernel<<<2048, 256, 0, stream>>>((const float4*)k_cache,
                                              (float4*)out, n4);
  copy_async_kernel<<<2048, 256, 0, stream>>>((const float4*)v_cache,
                                              (float4*)(out + vOffFloats), n4);

  // 3) scatter the new tokens (overwrites the freshly-copied zero blocks)
  const int nWaves = B_ * L_ * H_;    // 65,536 (token, head) pairs
  scatter_kernel<<<nWaves / 8, 256, 0, stream>>>((const float4*)key_states,
                                                 (const float4*)value_states,
                                                 cu_seqlens, seq_lens, bt_ws,
                                                 (float4*)out);
}